// PostProcessor_76931454206260
// MI455X (gfx1250) — compile-verified
//
#include <hip/hip_runtime.h>

#define K_B 2
#define K_N 512
#define K_C 151
#define K_D 4096
#define W_IMGF 800.0f
#define H_IMGF 600.0f
#define BBOX_CLAMP 4.135166556742356f
#define SCORE_THRESH 0.05f
#define NMS_T 0.5f
#define TOPN 300
#define DETPI 100

typedef __attribute__((ext_vector_type(2))) float v2f;
typedef __attribute__((ext_vector_type(8))) float v8f;

// ---------------- k1: softmax over classes + box decode + clip ----------------
__global__ __launch_bounds__(256) void k1_softmax_decode(
    const float* __restrict__ logits, const float* __restrict__ reg,
    const float* __restrict__ props, float* __restrict__ prob,
    float* __restrict__ boxes) {
  int bn = blockIdx.x;            // 0 .. B*N-1
  int c = threadIdx.x;            // 0 .. 255 (uses 0..150)
  __shared__ float red[256];
  float lg = (c < K_C) ? logits[bn * K_C + c] : -3.0e38f;
  red[c] = lg;
  __syncthreads();
  for (int s = 128; s > 0; s >>= 1) {
    if (c < s) red[c] = fmaxf(red[c], red[c + s]);
    __syncthreads();
  }
  float mx = red[0];
  __syncthreads();
  float e = (c < K_C) ? expf(lg - mx) : 0.0f;
  red[c] = e;
  __syncthreads();
  for (int s = 128; s > 0; s >>= 1) {
    if (c < s) red[c] += red[c + s];
    __syncthreads();
  }
  float p = e / red[0];
  if (c < K_C) {
    prob[bn * K_C + c] = p;
    const float* pb = props + bn * 4;
    float x1 = pb[0], y1 = pb[1], x2 = pb[2], y2 = pb[3];
    float w = x2 - x1 + 1.0f, h = y2 - y1 + 1.0f;
    float cx = x1 + 0.5f * w, cy = y1 + 0.5f * h;
    const float* rg = reg + (bn * K_C + c) * 4;
    float dx = rg[0] / 10.0f, dy = rg[1] / 10.0f;
    float dw = fminf(rg[2] / 5.0f, BBOX_CLAMP);
    float dh = fminf(rg[3] / 5.0f, BBOX_CLAMP);
    float pcx = dx * w + cx, pcy = dy * h + cy;
    float pw = expf(dw) * w, ph = expf(dh) * h;
    float bx0 = pcx - 0.5f * pw, by0 = pcy - 0.5f * ph;
    float bx1 = pcx + 0.5f * pw - 1.0f, by1 = pcy + 0.5f * ph - 1.0f;
    bx0 = fminf(fmaxf(bx0, 0.0f), W_IMGF - 1.0f);
    bx1 = fminf(fmaxf(bx1, 0.0f), W_IMGF - 1.0f);
    by0 = fminf(fmaxf(by0, 0.0f), H_IMGF - 1.0f);
    by1 = fminf(fmaxf(by1, 0.0f), H_IMGF - 1.0f);
    float* ob = boxes + (bn * K_C + c) * 4;
    ob[0] = bx0; ob[1] = by0; ob[2] = bx1; ob[3] = by1;
  }
}

// ---------------- k2: per-(image,class) NMS -----------------------------------
// One 512-thread block per (b, c). Bitonic argsort by (-score, idx) in LDS,
// then 16 waves build the 512x512 suppression bitmask in 16x16 tiles: the
// area_i + area_j term comes from V_WMMA_F32_16X16X4_F32 (exact fp32),
// intersections from VALU, predicates -> row bits via wave32 ballot, merged
// with ds_or_b32. The IoU>T test is done divide-free as
// inter > T*(area_sum - inter) (union strictly positive after clipping).
// A single lane then runs the serial greedy scan over the 16-word row masks
// and scatters kept scores back to original order.
__global__ __launch_bounds__(512) void k2_nms(
    const float* __restrict__ prob, const float* __restrict__ boxes,
    float* __restrict__ dist) {
  int blk = blockIdx.x;
  int b = blk / K_C;
  int c = blk % K_C;
  int t = threadIdx.x;
  if (c == 0) {  // background class: dropped
    dist[(b * K_N + t) * K_C + 0] = 0.0f;
    return;
  }

  __shared__ unsigned long long key[K_N];   // sort keys (desc score, asc idx)
  __shared__ float sbox[K_N * 4];           // sorted boxes
  __shared__ float sarea[K_N];              // sorted areas (+1 convention)
  __shared__ float sscore[K_N];             // sorted scores
  __shared__ int svalid[K_N];               // score > thresh prefilter
  __shared__ unsigned smask[K_N * 16];      // 512x512 suppression bitmask

  float sc = prob[(b * K_N + t) * K_C + c];
  unsigned fb = __float_as_uint(sc);
  unsigned ascn = fb ^ ((fb >> 31) ? 0xFFFFFFFFu : 0x80000000u);
  key[t] = ((unsigned long long)(~ascn) << 32) | (unsigned)t;
  __syncthreads();

  // bitonic sort ascending (=> descending score, ties -> ascending index)
  for (int k = 2; k <= K_N; k <<= 1) {
    for (int j = k >> 1; j > 0; j >>= 1) {
      int ixj = t ^ j;
      if (ixj > t) {
        unsigned long long a = key[t], bb = key[ixj];
        bool up = ((t & k) == 0);
        if ((a > bb) == up) { key[t] = bb; key[ixj] = a; }
      }
      __syncthreads();
    }
  }

  int orig = (int)(key[t] & 0xFFFFFFFFull);
  float s2 = prob[(b * K_N + orig) * K_C + c];
  sscore[t] = s2;
  svalid[t] = (s2 > SCORE_THRESH) ? 1 : 0;
  const float* bx = boxes + ((b * K_N + orig) * K_C + c) * 4;
  float b0 = bx[0], b1 = bx[1], b2 = bx[2], b3 = bx[3];
  sbox[t * 4 + 0] = b0; sbox[t * 4 + 1] = b1;
  sbox[t * 4 + 2] = b2; sbox[t * 4 + 3] = b3;
  sarea[t] = (b2 - b0 + 1.0f) * (b3 - b1 + 1.0f);
#pragma unroll
  for (int w = 0; w < 16; ++w) smask[t * 16 + w] = 0u;
  __syncthreads();

  int wave = t >> 5;
  int lane = t & 31;
  // 32x32 tiles of 16x16 -> 1024 tiles, 64 per wave
  for (int tile = wave; tile < 1024; tile += 16) {
    int ti = tile >> 5, tj = tile & 31;
    // A (16x4): col0 = area_i, col1 = 1  |  B (4x16): row0 = 1, row1 = area_j
    v2f av, bv;
    if (lane < 16) {
      av.x = sarea[ti * 16 + lane]; av.y = 1.0f;
      bv.x = 1.0f;                  bv.y = sarea[tj * 16 + lane];
    } else {
      av.x = 0.0f; av.y = 0.0f; bv.x = 0.0f; bv.y = 0.0f;
    }
    v8f c0 = {0.0f, 0.0f, 0.0f, 0.0f, 0.0f, 0.0f, 0.0f, 0.0f};
    // D[i][j] = area_i + area_j (exact fp32)
    v8f dsum = __builtin_amdgcn_wmma_f32_16x16x4_f32(
        false, av, false, bv, (short)0, c0, false, false);

    int jl = lane & 15;
    int j = tj * 16 + jl;
    float cb0 = sbox[j * 4 + 0], cb1 = sbox[j * 4 + 1];
    float cb2 = sbox[j * 4 + 2], cb3 = sbox[j * 4 + 3];
    unsigned hw = 0;
#pragma unroll
    for (int r = 0; r < 8; ++r) {
      int m = (lane < 16) ? r : (r + 8);      // D lane layout
      int i = ti * 16 + m;
      float r0 = sbox[i * 4 + 0], r1 = sbox[i * 4 + 1];
      float r2 = sbox[i * 4 + 2], r3 = sbox[i * 4 + 3];
      float xx1 = fmaxf(r0, cb0), yy1 = fmaxf(r1, cb1);
      float xx2 = fminf(r2, cb2), yy2 = fminf(r3, cb3);
      float inter = fmaxf(xx2 - xx1 + 1.0f, 0.0f) *
                    fmaxf(yy2 - yy1 + 1.0f, 0.0f);
      // iou > T  <=>  inter > T * (area_i + area_j - inter); union > 0 here
      bool sup = inter > NMS_T * (dsum[r] - inter);
      unsigned bal = (unsigned)__ballot(sup ? 1 : 0);
      // ballot: bits[15:0] = row r cols 0..15, bits[31:16] = row r+8 cols 0..15
      if (lane == r) hw = bal & 0xFFFFu;
      else if (lane == r + 8) hw = bal >> 16;
    }
    if (lane < 16) {
      unsigned word = hw << ((tj & 1) * 16);
      atomicOr(&smask[(ti * 16 + lane) * 16 + (tj >> 1)], word);
    }
  }
  __syncthreads();

  // serial greedy scan (sorted order), then cumsum<=300 cap
  if (t == 0) {
    unsigned acc[16];
#pragma unroll
    for (int w = 0; w < 16; ++w) acc[w] = 0u;
    int kept = 0;
    for (int i = 0; i < K_N; ++i) {
      int w = i >> 5, bbit = i & 31;
      bool active = svalid[i] && !((acc[w] >> bbit) & 1u);
      bool keepf = false;
      if (active) {
        kept++;
        keepf = (kept <= TOPN);
        unsigned lowmask = (bbit == 31) ? 0xFFFFFFFFu
                                        : ((1u << (bbit + 1)) - 1u);
        for (int w2 = w; w2 < 16; ++w2) {
          unsigned m = smask[i * 16 + w2];
          if (w2 == w) m &= ~lowmask;   // only suppress j > i
          acc[w2] |= m;
        }
      }
      int oi = (int)(key[i] & 0xFFFFFFFFull);
      dist[(b * K_N + oi) * K_C + c] = keepf ? sscore[i] : 0.0f;
    }
  }
}

// ---------------- k3: per-proposal argmax over classes ------------------------
__global__ void k3_argmax(const float* __restrict__ dist,
                          float* __restrict__ scores_pre,
                          int* __restrict__ labels) {
  int gid = blockIdx.x * blockDim.x + threadIdx.x;
  if (gid >= K_B * K_N) return;
  float best = -1.0f;
  int lab = 0;
  for (int c = 0; c < K_C; ++c) {
    float v = dist[gid * K_C + c];
    if (v > best) { best = v; lab = c; }   // strict > keeps first max
  }
  scores_pre[gid] = best;  // dist >= 0 so best >= 0
  labels[gid] = lab;
}

// ---------------- k4: per-image 100th-largest threshold -----------------------
__global__ __launch_bounds__(512) void k4_thresh(
    const float* __restrict__ scores_pre, float* __restrict__ thr) {
  int b = blockIdx.x;
  int t = threadIdx.x;
  __shared__ unsigned sk[K_N];
  __shared__ int cnt;
  if (t == 0) cnt = 0;
  float s = scores_pre[b * K_N + t];
  bool valid = s > 0.0f;
  float v = valid ? s : -1.0f;
  unsigned fb = __float_as_uint(v);
  unsigned ascn = fb ^ ((fb >> 31) ? 0xFFFFFFFFu : 0x80000000u);
  sk[t] = ~ascn;  // ascending sort of this key == descending value
  __syncthreads();
  if (valid) atomicAdd(&cnt, 1);
  for (int k = 2; k <= K_N; k <<= 1) {
    for (int j = k >> 1; j > 0; j >>= 1) {
      int ixj = t ^ j;
      if (ixj > t) {
        unsigned a = sk[t], bb = sk[ixj];
        bool up = ((t & k) == 0);
        if ((a > bb) == up) { sk[t] = bb; sk[ixj] = a; }
      }
      __syncthreads();
    }
  }
  if (t == 0) {
    unsigned ascv = ~sk[DETPI - 1];
    unsigned fbv = (ascv >> 31) ? (ascv ^ 0x80000000u) : ~ascv;
    float kth = __uint_as_float(fbv);
    thr[b] = (cnt > DETPI) ? kth : -1.0f;
  }
}

// ---------------- k5: finalize boxes/scores/labels/valid ----------------------
__global__ void k5_finalize(const float* __restrict__ scores_pre,
                            const int* __restrict__ labels,
                            const float* __restrict__ thr,
                            const float* __restrict__ boxes,
                            float* __restrict__ out,
                            float* __restrict__ detf) {
  int gid = blockIdx.x * blockDim.x + threadIdx.x;
  if (gid >= K_B * K_N) return;
  int b = gid >> 9;
  float s = scores_pre[gid];
  int lab = labels[gid];
  bool det = (s > 0.0f) && (s >= thr[b]);
  const float* bx = boxes + (gid * K_C + lab) * 4;
  float* ob = out + gid * 4;
  for (int k = 0; k < 4; ++k) ob[k] = det ? bx[k] : 0.0f;
  out[K_B * K_N * 4 + gid] = det ? s : 0.0f;           // final_scores
  out[K_B * K_N * 5 + gid] = det ? (float)lab : 0.0f;  // final_labels
  out[K_B * K_N * 6 + gid] = det ? 1.0f : 0.0f;        // det_valid
  detf[gid] = det ? 1.0f : 0.0f;
}

// ---------------- k6: feature masking (bandwidth-bound stream) ----------------
__global__ void k6_features(const float* __restrict__ feat,
                            const float* __restrict__ detf,
                            float* __restrict__ out) {
  int idx = blockIdx.x * blockDim.x + threadIdx.x;
  if (idx >= K_B * K_N * K_D) return;
  int bn = idx >> 12;  // D = 4096
  out[K_B * K_N * 7 + idx] = feat[idx] * detf[bn];
}

extern "C" void kernel_launch(void* const* d_in, const int* in_sizes, int n_in,
                              void* d_out, int out_size, void* d_ws, size_t ws_size,
                              hipStream_t stream) {
  const float* logits = (const float*)d_in[0];  // [2,512,151]
  const float* reg    = (const float*)d_in[1];  // [2,512,604]
  const float* props  = (const float*)d_in[2];  // [2,512,4]
  const float* feat   = (const float*)d_in[3];  // [2,512,4096]
  float* out = (float*)d_out;
  char* ws = (char*)d_ws;

  float* prob   = (float*)(ws + 0);        // 618,496 B  [B,N,C]
  float* boxes  = (float*)(ws + 618496);   // 2,473,984 B [B,N,C,4]
  float* dist   = (float*)(ws + 3092480);  // 618,496 B  [B,N,C]
  float* scores = (float*)(ws + 3710976);  // 4,096 B    [B,N]
  int*   labels = (int*)  (ws + 3715072);  // 4,096 B    [B,N]
  float* thr    = (float*)(ws + 3719168);  // 16 B       [B]
  float* detf   = (float*)(ws + 3719184);  // 4,096 B    [B,N]

  k1_softmax_decode<<<K_B * K_N, 256, 0, stream>>>(logits, reg, props, prob, boxes);
  k2_nms<<<K_B * K_C, 512, 0, stream>>>(prob, boxes, dist);
  k3_argmax<<<(K_B * K_N + 255) / 256, 256, 0, stream>>>(dist, scores, labels);
  k4_thresh<<<K_B, 512, 0, stream>>>(scores, thr);
  k5_finalize<<<(K_B * K_N + 255) / 256, 256, 0, stream>>>(scores, labels, thr,
                                                           boxes, out, detf);
  k6_features<<<(K_B * K_N * K_D + 255) / 256, 256, 0, stream>>>(feat, detf, out);
}